// TVL1OF_75118978007437
// MI455X (gfx1250) — compile-verified
//
#include <hip/hip_runtime.h>
#include <cmath>

// ---------------- problem constants (match reference) ----------------
#define B_N       4
#define H_N       256
#define W_N       256
#define HW_N      (H_N * W_N)
#define NUM_ITER  20
#define EPS_F     1e-8f

// ---------------- launch geometry ----------------
#define NBLOCKS         64          // 4 batches x 16 row-bands; co-resident (wave32, 8 waves/block)
#define NTHREADS        256         // one thread per column
#define RPB             16          // rows per block
#define BLKS_PER_BATCH  (H_N / RPB) // 16
#define LROWS           (RPB + 2)   // staged rows incl. halo
#define SEGS            (W_N / 4)   // 16B segments per row

// ---------------- CDNA5 async LDS path ----------------
#if defined(__HIP_DEVICE_COMPILE__) && defined(__gfx1250__) && \
    __has_builtin(__builtin_amdgcn_global_load_async_to_lds_b128)
#define TVL1_ASYNC 1
typedef int v4i_t __attribute__((ext_vector_type(4)));
typedef __attribute__((address_space(1))) v4i_t v4i_g;   // global
typedef __attribute__((address_space(3))) v4i_t v4i_l;   // LDS
#else
#define TVL1_ASYNC 0
#endif

// ---------------- device helpers ----------------
__device__ __forceinline__ void wait_async0() {
#if TVL1_ASYNC
#if __has_builtin(__builtin_amdgcn_s_wait_asynccnt)
  __builtin_amdgcn_s_wait_asynccnt(0);
#else
  asm volatile("s_wait_asynccnt 0" ::: "memory");
#endif
#endif
}

// Monotonic-counter grid barrier (cooperative persistent kernel).
__device__ __forceinline__ void grid_sync(unsigned* bar, unsigned target) {
  __threadfence();   // release: each wave flushes its own global writes to device scope
  __syncthreads();
  if (threadIdx.x == 0) {
    __hip_atomic_fetch_add(bar, 1u, __ATOMIC_RELAXED, __HIP_MEMORY_SCOPE_AGENT);
    while (__hip_atomic_load(bar, __ATOMIC_RELAXED, __HIP_MEMORY_SCOPE_AGENT) < target) {
      __builtin_amdgcn_s_sleep(2);
    }
  }
  __syncthreads();
  __threadfence();   // acquire: invalidate stale WGP-level cached lines
}

// Stage two full-width plane tiles (rows row0-1 .. row0+RPB) into LDS.
// sp layout: [plane][LROWS][W_N]. Out-of-image halo rows are zero-filled (SAME padding).
__device__ __forceinline__ void stage2(const float* __restrict__ pa,
                                       const float* __restrict__ pb,
                                       float* sp, int row0) {
  const int tid = threadIdx.x;
  const int total = 2 * LROWS * SEGS;  // 2304 -> 9 per thread
#pragma unroll 1
  for (int t = tid; t < total; t += NTHREADS) {
    const int plane = t / (LROWS * SEGS);
    const int rem   = t - plane * (LROWS * SEGS);
    const int rrow  = rem / SEGS;
    const int seg   = rem - rrow * SEGS;
    const int gy    = row0 - 1 + rrow;
    float* dst = sp + (size_t)(plane * LROWS + rrow) * W_N + seg * 4;
    if ((unsigned)gy < (unsigned)H_N) {
      const float* g = (plane ? pb : pa) + (size_t)gy * W_N + seg * 4;
#if TVL1_ASYNC
      __builtin_amdgcn_global_load_async_to_lds_b128((v4i_g*)g, (v4i_l*)dst, 0, 0);
#else
      const float4 v = *(const float4*)g;
      *(float4*)dst = v;
#endif
    } else {
      float4 z; z.x = z.y = z.z = z.w = 0.f;
      *(float4*)dst = z;
    }
  }
  wait_async0();
  __syncthreads();
}

__device__ __forceinline__ float ldsv(const float* sp, int plane, int r, int xx) {
  float v = 0.f;
  if ((unsigned)xx < (unsigned)W_N) v = sp[(size_t)(plane * LROWS + r) * W_N + xx];
  return v;
}

// cross-correlation with SX = [[-1,0,1],[-2,0,2],[-1,0,1]] ; LDS row i == image row y-1
__device__ __forceinline__ float convSX_l(const float* sp, int plane, int i, int x) {
  return (ldsv(sp, plane, i + 0, x + 1) - ldsv(sp, plane, i + 0, x - 1)) +
         2.f * (ldsv(sp, plane, i + 1, x + 1) - ldsv(sp, plane, i + 1, x - 1)) +
         (ldsv(sp, plane, i + 2, x + 1) - ldsv(sp, plane, i + 2, x - 1));
}
// cross-correlation with SY = [[-1,-2,-1],[0,0,0],[1,2,1]]
__device__ __forceinline__ float convSY_l(const float* sp, int plane, int i, int x) {
  return (ldsv(sp, plane, i + 2, x - 1) - ldsv(sp, plane, i + 0, x - 1)) +
         2.f * (ldsv(sp, plane, i + 2, x) - ldsv(sp, plane, i + 0, x)) +
         (ldsv(sp, plane, i + 2, x + 1) - ldsv(sp, plane, i + 0, x + 1));
}

__device__ __forceinline__ float gat(const float* p, int y, int x) {
  if ((unsigned)y >= (unsigned)H_N || (unsigned)x >= (unsigned)W_N) return 0.f;
  return p[(size_t)y * W_N + x];
}

// ---------------- kernels ----------------
__global__ void tvl1_init_bar(unsigned* bar) {
  if (threadIdx.x == 0) *bar = 0u;
}

__global__ void __launch_bounds__(NTHREADS)
tvl1_persistent(const float* __restrict__ x,
                const float* __restrict__ lam_p,
                const float* __restrict__ tau_p,
                const float* __restrict__ theta_p,
                float* __restrict__ u,       // d_out, layout (B,2,H,W)
                float* __restrict__ ws,      // scratch planes
                unsigned* bar) {
  __shared__ __align__(16) float sp[2 * LROWS * W_N];  // 36 KB of the 320 KB WGP LDS

  const int tid  = threadIdx.x;          // column index
  const int blk  = blockIdx.x;
  const int b    = blk / BLKS_PER_BATCH;
  const int row0 = (blk % BLKS_PER_BATCH) * RPB;
  const int xc   = tid;

  float* p1    = ws;                                   // B*2*HW
  float* p2    = ws + (size_t)2 * B_N * HW_N;          // B*2*HW
  float* rho_c = ws + (size_t)4 * B_N * HW_N;          // B*HW
  float* gxp   = rho_c + (size_t)B_N * HW_N;
  float* gyp   = gxp + (size_t)B_N * HW_N;
  float* ngp   = gyp + (size_t)B_N * HW_N;

  const float lambda_ = lam_p[0];
  const float tau     = tau_p[0];
  const float theta   = theta_p[0];
  const float r       = tau / theta;
  const float tl      = theta * lambda_;

  const float* im0 = x + (size_t)(b * 2 + 0) * HW_N;
  const float* im1 = x + (size_t)(b * 2 + 1) * HW_N;

  // ---------- init: rho_c, grad_im, norm_grad; zero u, p1, p2 ----------
#pragma unroll
  for (int i = 0; i < RPB; ++i) {
    const int y = row0 + i;
    const float gx =
        (gat(im1, y - 1, xc + 1) - gat(im1, y - 1, xc - 1)) +
        2.f * (gat(im1, y, xc + 1) - gat(im1, y, xc - 1)) +
        (gat(im1, y + 1, xc + 1) - gat(im1, y + 1, xc - 1));
    const float gy =
        (gat(im1, y + 1, xc - 1) - gat(im1, y - 1, xc - 1)) +
        2.f * (gat(im1, y + 1, xc) - gat(im1, y - 1, xc)) +
        (gat(im1, y + 1, xc + 1) - gat(im1, y - 1, xc + 1));
    const size_t idx = (size_t)b * HW_N + (size_t)y * W_N + xc;
    gxp[idx] = gx;
    gyp[idx] = gy;
    ngp[idx] = gx * gx + gy * gy + EPS_F;
    rho_c[idx] = im1[(size_t)y * W_N + xc] - im0[(size_t)y * W_N + xc];
    for (int c = 0; c < 2; ++c) {
      const size_t pidx = (size_t)(b * 2 + c) * HW_N + (size_t)y * W_N + xc;
      u[pidx] = 0.f;
      p1[pidx] = 0.f;
      p2[pidx] = 0.f;
    }
  }

  unsigned gen = 1;
  grid_sync(bar, gen * NBLOCKS);

  const size_t ub0 = (size_t)(b * 2 + 0) * HW_N;
  const size_t ub1 = (size_t)(b * 2 + 1) * HW_N;

  for (int it = 0; it < NUM_ITER; ++it) {
    // ================= Phase A: u update =================
    float dval[RPB];

    // channel 0: needs convSX(p1_0) + convSY(p2_0)
    stage2(p1 + ub0, p2 + ub0, sp, row0);
#pragma unroll
    for (int i = 0; i < RPB; ++i) {
      const int y = row0 + i;
      const size_t idx = (size_t)b * HW_N + (size_t)y * W_N + xc;
      const size_t o = (size_t)y * W_N + xc;
      const float u0 = u[ub0 + o];
      const float u1 = u[ub1 + o];
      const float gx = gxp[idx], gy = gyp[idx], ng = ngp[idx];
      const float rho = rho_c[idx] + gx * u0 + gy * u1;
      const float th = tl * ng;
      const float d = (fabsf(rho) < th) ? (rho / ng) : copysignf(tl, rho);
      dval[i] = d;
      const float div0 = convSX_l(sp, 0, i, xc) + convSY_l(sp, 1, i, xc);
      u[ub0 + o] = u0 - d * gx + theta * div0;
    }
    __syncthreads();  // all reads of sp done before restage

    // channel 1: needs convSX(p1_1) + convSY(p2_1)
    stage2(p1 + ub1, p2 + ub1, sp, row0);
#pragma unroll
    for (int i = 0; i < RPB; ++i) {
      const int y = row0 + i;
      const size_t idx = (size_t)b * HW_N + (size_t)y * W_N + xc;
      const size_t o = (size_t)y * W_N + xc;
      const float u1 = u[ub1 + o];
      const float gy = gyp[idx];
      const float div1 = convSX_l(sp, 0, i, xc) + convSY_l(sp, 1, i, xc);
      u[ub1 + o] = u1 - dval[i] * gy + theta * div1;
    }
    gen++;
    grid_sync(bar, gen * NBLOCKS);  // u consistent device-wide

    // ================= Phase B: p update =================
    stage2(u + ub0, u + ub1, sp, row0);
#pragma unroll
    for (int i = 0; i < RPB; ++i) {
      const int y = row0 + i;
      const size_t o = (size_t)y * W_N + xc;
      const float g1x = convSX_l(sp, 0, i, xc);
      const float g1y = convSY_l(sp, 0, i, xc);
      const float g2x = convSX_l(sp, 1, i, xc);
      const float g2y = convSY_l(sp, 1, i, xc);
      const float den1 = 1.f + r * (fabsf(g1x) + fabsf(g1y));
      const float den2 = 1.f + r * (fabsf(g2x) + fabsf(g2y));
      p1[ub0 + o] = (p1[ub0 + o] + r * g1x) / den1;
      p1[ub1 + o] = (p1[ub1 + o] + r * g1y) / den1;
      p2[ub0 + o] = (p2[ub0 + o] + r * g2x) / den2;
      p2[ub1 + o] = (p2[ub1 + o] + r * g2y) / den2;
    }
    gen++;
    grid_sync(bar, gen * NBLOCKS);  // p consistent before next iteration
  }
  // u already lives in d_out
}

// ---------------- host entry ----------------
extern "C" void kernel_launch(void* const* d_in, const int* in_sizes, int n_in,
                              void* d_out, int out_size, void* d_ws, size_t ws_size,
                              hipStream_t stream) {
  (void)in_sizes; (void)n_in; (void)out_size; (void)ws_size;
  const float* x      = (const float*)d_in[0];
  const float* lam_p  = (const float*)d_in[1];
  const float* tau_p  = (const float*)d_in[2];
  const float* theta_p= (const float*)d_in[3];
  float* u  = (float*)d_out;
  float* ws = (float*)d_ws;
  // plane scratch uses 8 MiB; barrier counter right after
  unsigned* bar = (unsigned*)((char*)d_ws + (size_t)8 * 1024 * 1024);

  tvl1_init_bar<<<1, 32, 0, stream>>>(bar);
  tvl1_persistent<<<NBLOCKS, NTHREADS, 0, stream>>>(x, lam_p, tau_p, theta_p, u, ws, bar);
}